// LinearTransformerCausalDiagonalDecoder_42691974922504
// MI455X (gfx1250) — compile-verified
//
#include <hip/hip_runtime.h>
#include <math.h>

// ---------------- CDNA5 WMMA fp32 16x16x4 ----------------
typedef __attribute__((ext_vector_type(2))) float v2f;
typedef __attribute__((ext_vector_type(8))) float v8f;
typedef __attribute__((ext_vector_type(4))) int   v4i;

__device__ __forceinline__ v8f wmma_f32x4(v2f a, v2f b, v8f c) {
  // D = A(16x4) * B(4x16) + C(16x16), fp32, wave32
  return __builtin_amdgcn_wmma_f32_16x16x4_f32(false, a, false, b, (short)0, c, false, false);
}

// ---------------- gfx1250 async global->LDS path ----------------
#if defined(__has_builtin)
#if __has_builtin(__builtin_amdgcn_global_load_async_to_lds_b128) && \
    __has_builtin(__builtin_amdgcn_s_wait_asynccnt)
#define USE_ASYNC_LDS 1
#endif
#endif
#ifndef USE_ASYNC_LDS
#define USE_ASYNC_LDS 0
#endif

#if USE_ASYNC_LDS
__device__ __forceinline__ void cp16_async(const float* g, float* l) {
  // prototype (leaked by clang diagnostic): (v4i AS1*, v4i AS3*, imm offset, imm cpol)
  __builtin_amdgcn_global_load_async_to_lds_b128(
      (__attribute__((address_space(1))) v4i*)g,
      (__attribute__((address_space(3))) v4i*)l, 0, 0);
}
#endif

#define D_MODEL 512
#define D_FF    2048
#define NTOK    2048   // B*S
#define SEQ     1024
#define NHEAD   8
#define HDIM    64
#define EPS_LN  1e-5f
#define EPS_AT  1e-6f

__device__ __forceinline__ float gelu_tanh(float x) {
  const float k0 = 0.7978845608028654f;  // sqrt(2/pi)
  const float k1 = 0.044715f;
  float x3 = x * x * x;
  return 0.5f * x * (1.0f + tanhf(k0 * (x + k1 * x3)));
}

__device__ __forceinline__ float phi_elu1(float x) {
  // elu(x)+1 : x>0 ? x+1 : exp(x)
  return (x > 0.0f) ? (x + 1.0f) : __expf(x);
}

// ---------------- GEMM: C[M,N] = A[M,K] @ B[K,N] + bias, optional GELU ----
// block = 256 threads = 8 waves; block tile 64(M) x 128(N); k-block 32.
// A tile staged in LDS at stride 36 (conflict-free K-pair row reads),
// B tile at stride 132. Waves: (wave&3) -> 16-row strip, (wave>>2) -> 64-col strip.
#define ASTR 36
#define BSTR 132

template <int GELU>
__global__ void gemm_bias_kernel(const float* __restrict__ A,
                                 const float* __restrict__ B,
                                 const float* __restrict__ bias,
                                 float* __restrict__ C,
                                 int M, int N, int K) {
  __shared__ __align__(16) float As[64 * ASTR];
  __shared__ __align__(16) float Bs[32 * BSTR];

  const int wave = threadIdx.x >> 5;
  const int lane = threadIdx.x & 31;
  const int lr = lane & 15;
  const int hi = lane >> 4;
  const int mw = (wave & 3) * 16;     // wave row offset inside block tile
  const int nw = (wave >> 2) * 64;    // wave col offset inside block tile

  const int mblk = blockIdx.x * 64;
  const int nblk = blockIdx.y * 128;

  v8f acc[4];
#pragma unroll
  for (int t = 0; t < 4; ++t) acc[t] = (v8f){0,0,0,0,0,0,0,0};

  for (int kb = 0; kb < K; kb += 32) {
    // ---- stage A(64x32) and B(32x128) tiles into LDS ----
    for (int i = threadIdx.x; i < 64 * 8; i += 256) {
      const int r = i >> 3, c = (i & 7) << 2;
      const float* gp = A + (size_t)(mblk + r) * K + kb + c;
      float* lp = &As[r * ASTR + c];
#if USE_ASYNC_LDS
      cp16_async(gp, lp);
#else
      *(float4*)lp = *(const float4*)gp;
#endif
    }
    for (int i = threadIdx.x; i < 32 * 32; i += 256) {
      const int r = i >> 5, c = (i & 31) << 2;
      const float* gp = B + (size_t)(kb + r) * N + nblk + c;
      float* lp = &Bs[r * BSTR + c];
#if USE_ASYNC_LDS
      cp16_async(gp, lp);
#else
      *(float4*)lp = *(const float4*)gp;
#endif
    }
#if USE_ASYNC_LDS
    __builtin_amdgcn_s_wait_asynccnt(0);
#endif
    __syncthreads();

    // ---- 8 WMMA K-steps x 4 N-tiles from LDS ----
#pragma unroll
    for (int kk = 0; kk < 32; kk += 4) {
      v2f a;
      a.x = As[(mw + lr) * ASTR + kk + 2 * hi];
      a.y = As[(mw + lr) * ASTR + kk + 2 * hi + 1];
#pragma unroll
      for (int t = 0; t < 4; ++t) {
        v2f b;
        b.x = Bs[(kk + 2 * hi) * BSTR + nw + t * 16 + lr];
        b.y = Bs[(kk + 2 * hi + 1) * BSTR + nw + t * 16 + lr];
        acc[t] = wmma_f32x4(a, b, acc[t]);
      }
    }
    __syncthreads();
  }

#pragma unroll
  for (int t = 0; t < 4; ++t) {
    const int col = nblk + nw + t * 16 + lr;
    const float bv = bias[col];
#pragma unroll
    for (int j = 0; j < 8; ++j) {
      const int row = mblk + mw + j + 8 * hi;
      float v = acc[t][j] + bv;
      if (GELU) v = gelu_tanh(v);
      C[(size_t)row * N + col] = v;
    }
  }
}

// ---------------- fused (x [+ r]) -> LayerNorm -------------------------
// one block (256 threads) per row of 512
__global__ void add_ln_kernel(const float* __restrict__ x,
                              const float* __restrict__ r,
                              const float* __restrict__ g,
                              const float* __restrict__ b,
                              float* __restrict__ out, int has_res) {
  const int row = blockIdx.x;
  const int t = threadIdx.x;
  const float* xr = x + (size_t)row * D_MODEL;
  float v0 = xr[t];
  float v1 = xr[t + 256];
  if (has_res) {
    const float* rr = r + (size_t)row * D_MODEL;
    v0 += rr[t];
    v1 += rr[t + 256];
  }
  __shared__ float red[8];
  __shared__ float red2[8];
  const int wv = t >> 5, ln = t & 31;

  float s = v0 + v1;
#pragma unroll
  for (int off = 16; off > 0; off >>= 1) s += __shfl_down(s, off, 32);
  if (ln == 0) red[wv] = s;
  __syncthreads();
  if (t == 0) {
    float tot = 0.f;
#pragma unroll
    for (int i = 0; i < 8; ++i) tot += red[i];
    red[0] = tot * (1.0f / (float)D_MODEL);
  }
  __syncthreads();
  const float mu = red[0];

  const float d0 = v0 - mu, d1 = v1 - mu;
  float q = d0 * d0 + d1 * d1;
#pragma unroll
  for (int off = 16; off > 0; off >>= 1) q += __shfl_down(q, off, 32);
  if (ln == 0) red2[wv] = q;
  __syncthreads();
  if (t == 0) {
    float tot = 0.f;
#pragma unroll
    for (int i = 0; i < 8; ++i) tot += red2[i];
    red2[0] = rsqrtf(tot * (1.0f / (float)D_MODEL) + EPS_LN);
  }
  __syncthreads();
  const float is = red2[0];

  float* orow = out + (size_t)row * D_MODEL;
  orow[t]       = d0 * is * g[t]       + b[t];
  orow[t + 256] = d1 * is * g[t + 256] + b[t + 256];
}

// ---------------- causal linear attention, chunked WMMA scan ------------
// grid = B*NHEAD blocks; block = 256 threads (8 waves); chunk = 64 rows.
// Q,K,V,O: [NTOK, 512] row-major, head h at cols h*64 .. h*64+63.
#define LSTR 65  // LDS row stride (bank-conflict padding)

__global__ void linattn_kernel(const float* __restrict__ Q,
                               const float* __restrict__ K,
                               const float* __restrict__ V,
                               float* __restrict__ O) {
  extern __shared__ float smem[];
  float* st   = smem;                 // 64x65  running state S[d][e]
  float* sq   = st + 64 * LSTR;       // 64x65  phi(Q) chunk
  float* sk   = sq + 64 * LSTR;       // 64x65  phi(K) chunk
  float* sv   = sk + 64 * LSTR;       // 64x65  V chunk
  float* sa   = sv + 64 * LSTR;       // 64x65  tril(Q K^T)
  float* kcum = sa + 64 * LSTR;       // 64     running sum of phi(K)
  float* zz   = kcum + 64;            // 64     denominators

  const int bh = blockIdx.x;          // 0..15
  const int bb = bh >> 3;
  const int hh = bh & 7;
  const size_t base = (size_t)bb * SEQ * D_MODEL + (size_t)hh * HDIM;

  const int tid = threadIdx.x;
  const int wave = tid >> 5, lane = tid & 31;
  const int lr = lane & 15, hi = lane >> 4;

  for (int i = tid; i < 64 * LSTR; i += 256) st[i] = 0.0f;
  if (tid < 64) kcum[tid] = 0.0f;
  __syncthreads();

  for (int c = 0; c < SEQ / 64; ++c) {
    // ---- 1. load chunk, apply phi to Q,K; prefetch next chunk ----
    for (int idx = tid; idx < 64 * 64; idx += 256) {
      const int r = idx >> 6, cc = idx & 63;
      const size_t g = base + (size_t)(c * 64 + r) * D_MODEL + cc;
      sq[r * LSTR + cc] = phi_elu1(Q[g]);
      sk[r * LSTR + cc] = phi_elu1(K[g]);
      sv[r * LSTR + cc] = V[g];
      if (c + 1 < SEQ / 64) {
        const size_t gn = g + (size_t)64 * D_MODEL;
        __builtin_prefetch(&Q[gn], 0, 0);
        __builtin_prefetch(&K[gn], 0, 0);
        __builtin_prefetch(&V[gn], 0, 0);
      }
    }
    __syncthreads();

    // ---- 2. A = tril(Qc @ Kc^T); 16 tiles, 2 per wave ----
#pragma unroll
    for (int t2 = 0; t2 < 2; ++t2) {
      const int tile = wave * 2 + t2;
      const int mt = tile >> 2, nt = tile & 3;
      v8f acc = (v8f){0,0,0,0,0,0,0,0};
      for (int kk = 0; kk < 64; kk += 4) {
        v2f a, b;
        a.x = sq[(mt * 16 + lr) * LSTR + kk + 2 * hi];
        a.y = sq[(mt * 16 + lr) * LSTR + kk + 2 * hi + 1];
        b.x = sk[(nt * 16 + lr) * LSTR + kk + 2 * hi];       // B[k][j] = K[j][k]
        b.y = sk[(nt * 16 + lr) * LSTR + kk + 2 * hi + 1];
        acc = wmma_f32x4(a, b, acc);
      }
#pragma unroll
      for (int j = 0; j < 8; ++j) {
        const int i = mt * 16 + j + 8 * hi;
        const int jj = nt * 16 + lr;
        sa[i * LSTR + jj] = (jj <= i) ? acc[j] : 0.0f;
      }
    }
    __syncthreads();

    // ---- 3. z_i = q_i . kcum_prev + rowsum(tril A) ----
    if (tid < 64) {
      float s = 0.0f;
      for (int d = 0; d < 64; ++d) s += sq[tid * LSTR + d] * kcum[d];
      for (int j = 0; j < 64; ++j) s += sa[tid * LSTR + j];
      zz[tid] = s;
    }
    __syncthreads();

    // ---- 4. num = Qc@S_prev + A@Vc ; write O = num / (z+eps) ----
#pragma unroll
    for (int t2 = 0; t2 < 2; ++t2) {
      const int tile = wave * 2 + t2;
      const int mtr = tile >> 2, etc_ = tile & 3;
      v8f acc = (v8f){0,0,0,0,0,0,0,0};
      for (int kk = 0; kk < 64; kk += 4) {          // Qc @ S_prev
        v2f a, b;
        a.x = sq[(mtr * 16 + lr) * LSTR + kk + 2 * hi];
        a.y = sq[(mtr * 16 + lr) * LSTR + kk + 2 * hi + 1];
        b.x = st[(kk + 2 * hi) * LSTR + etc_ * 16 + lr];
        b.y = st[(kk + 2 * hi + 1) * LSTR + etc_ * 16 + lr];
        acc = wmma_f32x4(a, b, acc);
      }
      for (int kk = 0; kk < 64; kk += 4) {          // A @ Vc
        v2f a, b;
        a.x = sa[(mtr * 16 + lr) * LSTR + kk + 2 * hi];
        a.y = sa[(mtr * 16 + lr) * LSTR + kk + 2 * hi + 1];
        b.x = sv[(kk + 2 * hi) * LSTR + etc_ * 16 + lr];
        b.y = sv[(kk + 2 * hi + 1) * LSTR + etc_ * 16 + lr];
        acc = wmma_f32x4(a, b, acc);
      }
#pragma unroll
      for (int j = 0; j < 8; ++j) {
        const int il = mtr * 16 + j + 8 * hi;
        const int e = etc_ * 16 + lr;
        const size_t g = base + (size_t)(c * 64 + il) * D_MODEL + e;
        O[g] = acc[j] / (zz[il] + EPS_AT);
      }
    }
    __syncthreads();

    // ---- 5. S += Kc^T @ Vc ; kcum += colsum(Kc) ----
#pragma unroll
    for (int t2 = 0; t2 < 2; ++t2) {
      const int tile = wave * 2 + t2;
      const int dt = tile >> 2, et = tile & 3;
      v8f acc;
#pragma unroll
      for (int j = 0; j < 8; ++j)
        acc[j] = st[(dt * 16 + j + 8 * hi) * LSTR + et * 16 + lr];
      for (int jj = 0; jj < 64; jj += 4) {
        v2f a, b;
        a.x = sk[(jj + 2 * hi) * LSTR + dt * 16 + lr];       // A[d][j] = K[j][d]
        a.y = sk[(jj + 2 * hi + 1) * LSTR + dt * 16 + lr];
        b.x = sv[(jj + 2 * hi) * LSTR + et * 16 + lr];
        b.y = sv[(jj + 2 * hi + 1) * LSTR + et * 16 + lr];
        acc = wmma_f32x4(a, b, acc);
      }
#pragma unroll
      for (int j = 0; j < 8; ++j)
        st[(dt * 16 + j + 8 * hi) * LSTR + et * 16 + lr] = acc[j];
    }
    if (tid < 64) {
      float s = 0.0f;
      for (int j = 0; j < 64; ++j) s += sk[j * LSTR + tid];
      kcum[tid] += s;
    }
    __syncthreads();
  }
}

// ---------------- host orchestration ------------------------------------
extern "C" void kernel_launch(void* const* d_in, const int* in_sizes, int n_in,
                              void* d_out, int out_size, void* d_ws, size_t ws_size,
                              hipStream_t stream) {
  (void)in_sizes; (void)n_in; (void)out_size; (void)ws_size;
  const float* memory = (const float*)d_in[0];
  const float* target = (const float*)d_in[1];
  const float* Wq = (const float*)d_in[2];  const float* bq = (const float*)d_in[3];
  const float* Wk = (const float*)d_in[4];  const float* bk = (const float*)d_in[5];
  const float* Wv = (const float*)d_in[6];  const float* bv = (const float*)d_in[7];
  const float* Wo = (const float*)d_in[8];  const float* bo = (const float*)d_in[9];
  const float* Wvc = (const float*)d_in[10]; const float* bvc = (const float*)d_in[11];
  const float* Woc = (const float*)d_in[12]; const float* boc = (const float*)d_in[13];
  const float* W1 = (const float*)d_in[14]; const float* b1 = (const float*)d_in[15];
  const float* W2 = (const float*)d_in[16]; const float* b2 = (const float*)d_in[17];
  const float* ln1g = (const float*)d_in[18]; const float* ln1b = (const float*)d_in[19];
  const float* ln2g = (const float*)d_in[20]; const float* ln2b = (const float*)d_in[21];
  const float* ln3g = (const float*)d_in[22]; const float* ln3b = (const float*)d_in[23];
  const float* lnfg = (const float*)d_in[24]; const float* lnfb = (const float*)d_in[25];

  float* ws = (float*)d_ws;
  const size_t T = NTOK;
  float* xb = ws;               // T*512
  float* qb = xb + T * D_MODEL;
  float* kb = qb + T * D_MODEL;
  float* vb = kb + T * D_MODEL;
  float* ab = vb + T * D_MODEL;
  float* tb = ab + T * D_MODEL;
  float* t2 = tb + T * D_MODEL;
  float* hb = t2 + T * D_MODEL; // T*2048

  (void)hipMemcpyAsync(xb, target, T * D_MODEL * sizeof(float),
                       hipMemcpyDeviceToDevice, stream);

  const size_t smem_attn = (size_t)(5 * 64 * LSTR + 128) * sizeof(float);

  auto gemm = [&](const float* A, const float* B, const float* bias, float* C,
                  int M, int N, int Kd, bool gelu) {
    dim3 grid(M / 64, N / 128);
    if (gelu)
      gemm_bias_kernel<1><<<grid, 256, 0, stream>>>(A, B, bias, C, M, N, Kd);
    else
      gemm_bias_kernel<0><<<grid, 256, 0, stream>>>(A, B, bias, C, M, N, Kd);
  };

  for (int l = 0; l < 4; ++l) {
    const size_t wdd = (size_t)l * D_MODEL * D_MODEL;
    const size_t wdf = (size_t)l * D_MODEL * D_FF;
    const size_t vd  = (size_t)l * D_MODEL;
    const size_t vf  = (size_t)l * D_FF;

    // ---- causal linear self-attention ----
    gemm(xb, Wq + wdd, bq + vd, qb, NTOK, D_MODEL, D_MODEL, false);
    gemm(xb, Wk + wdd, bk + vd, kb, NTOK, D_MODEL, D_MODEL, false);
    gemm(xb, Wv + wdd, bv + vd, vb, NTOK, D_MODEL, D_MODEL, false);
    linattn_kernel<<<16, 256, smem_attn, stream>>>(qb, kb, vb, ab);
    gemm(ab, Wo + wdd, bo + vd, tb, NTOK, D_MODEL, D_MODEL, false);
    add_ln_kernel<<<NTOK, 256, 0, stream>>>(xb, tb, ln1g + vd, ln1b + vd, xb, 1);

    // ---- diagonal cross-attention ----
    gemm(memory, Wvc + wdd, bvc + vd, t2, NTOK, D_MODEL, D_MODEL, false);
    gemm(t2, Woc + wdd, boc + vd, tb, NTOK, D_MODEL, D_MODEL, false);
    add_ln_kernel<<<NTOK, 256, 0, stream>>>(xb, tb, ln2g + vd, ln2b + vd, xb, 1);

    // ---- gelu FFN ----
    gemm(xb, W1 + wdf, b1 + vf, hb, NTOK, D_FF, D_MODEL, true);
    gemm(hb, W2 + wdf, b2 + vd, tb, NTOK, D_MODEL, D_FF, false);
    add_ln_kernel<<<NTOK, 256, 0, stream>>>(xb, tb, ln3g + vd, ln3b + vd, xb, 1);
  }

  // ---- final layer norm into output ----
  add_ln_kernel<<<NTOK, 256, 0, stream>>>(xb, nullptr, lnfg, lnfb, (float*)d_out, 0);
}